// BahdanauAttention_13211319403256
// MI455X (gfx1250) — compile-verified
//
#include <hip/hip_runtime.h>
#include <hip/hip_bf16.h>
#include <math.h>

// ---------------------------------------------------------------------------
// Bahdanau attention, MI455X (gfx1250, wave32).
//   B=64, S=2048, Q=512
//   pq      = query @ Wq^T                     (WMMA f32 16x16x4)
//   energy  = sum_q We[q]*tanh(pq + PM)        (bandwidth bound: 256MB stream)
//   attn    = softmax(mask ? energy : -1000)   (per-batch, tiny)
// ---------------------------------------------------------------------------

#define BQ   64
#define SQ   2048
#define QQ   512

typedef __attribute__((ext_vector_type(2))) float v2f;
typedef __attribute__((ext_vector_type(4))) float fv4;
typedef __attribute__((ext_vector_type(8))) float v8f;

// gfx1250 hardware tanh (TRANS op) if the toolchain exposes it.
#if __has_builtin(__builtin_amdgcn_tanhf)
  #define FAST_TANH(x) __builtin_amdgcn_tanhf(x)
#elif __has_builtin(__builtin_amdgcn_tanh_f32)
  #define FAST_TANH(x) __builtin_amdgcn_tanh_f32(x)
#else
  #define FAST_TANH(x) tanhf(x)
#endif

// ---------------------------------------------------------------------------
// Kernel 1: pq[m, n] = sum_k query[m, k] * Wq[n, k]
// M=64 (batch), N=512 (d), K=512 (q). One wave per 16x16 tile, K stepped by 4
// via V_WMMA_F32_16X16X4_F32.
//
// ISA 32-bit A (16x4, MxK) layout: lane&15 = M row; VGPR j + 2*(lane>>4) = K.
// B (4x16, KxN) mirrors with lane&15 = N col.
// C/D f32 layout: VGPR r, lane l -> D[r + 8*(l>>4)][l&15].
// ---------------------------------------------------------------------------
__global__ __launch_bounds__(32)
void proj_query_wmma(const float* __restrict__ query,   // [B, Q] (t=1 squeezed)
                     const float* __restrict__ Wq,      // [Q, Q] row-major
                     float* __restrict__ pq)            // [B, Q]
{
    const int lane = threadIdx.x;          // 0..31
    const int n0   = blockIdx.x * 16;      // 32 tiles over N=512
    const int m0   = blockIdx.y * 16;      // 4 tiles over M=64
    const int half = lane >> 4;            // 0/1
    const int lid  = lane & 15;

    const float* aRow = query + (size_t)(m0 + lid) * QQ;  // A: row m0+lid
    const float* bRow = Wq    + (size_t)(n0 + lid) * QQ;  // B[k][n]=Wq[n][k]

    v8f acc = {};
    #pragma unroll 4
    for (int k = 0; k < QQ; k += 4) {
        const int kk = k + 2 * half;
        v2f a, b;
        a.x = aRow[kk + 0];
        a.y = aRow[kk + 1];
        b.x = bRow[kk + 0];
        b.y = bRow[kk + 1];
        acc = __builtin_amdgcn_wmma_f32_16x16x4_f32(
                  /*neg_a=*/false, a, /*neg_b=*/false, b,
                  /*c_mod=*/(short)0, acc,
                  /*reuse_a=*/false, /*reuse_b=*/false);
    }

    #pragma unroll
    for (int r = 0; r < 8; ++r) {
        pq[(size_t)(m0 + r + 8 * half) * QQ + (n0 + lid)] = acc[r];
    }
}

// ---------------------------------------------------------------------------
// Kernel 2: energy[b,s] = sum_q We[q]*tanh(pq[b,q] + PM[b,s,q]); then mask.
// One wave per (b,s) row; 8 waves / block (all rows in a block share b, since
// 2048 % 8 == 0). Lane reads float4; 4 iterations cover Q=512, each iteration
// is a fully-coalesced 512B wave access. PM is streamed with non-temporal
// loads (single use, 256MB > L2).
// ---------------------------------------------------------------------------
__global__ __launch_bounds__(256)
void energy_kernel(const float* __restrict__ pm,     // [B, S, Q]
                   const int*   __restrict__ mask,   // [B, S]
                   const float* __restrict__ pq,     // [B, Q]
                   const float* __restrict__ We,     // [Q]
                   float* __restrict__ energy)       // [B*S]
{
    const int wave = threadIdx.x >> 5;
    const int lane = threadIdx.x & 31;
    const size_t row = (size_t)blockIdx.x * 8 + wave;   // row = b*S + s
    const int b = (int)(row >> 11);                     // S = 2048 = 2^11

    const float* pmRow = pm + row * QQ;
    const float* pqRow = pq + (size_t)b * QQ;

    float sum = 0.0f;
    #pragma unroll
    for (int i = 0; i < 4; ++i) {
        const int q = i * 128 + lane * 4;
        fv4 x = __builtin_nontemporal_load((const fv4*)(pmRow + q));
        fv4 p = *(const fv4*)(pqRow + q);   // hot in L1/L2 (128KB total)
        fv4 w = *(const fv4*)(We + q);      // hot (2KB)
        sum += w.x * FAST_TANH(p.x + x.x);
        sum += w.y * FAST_TANH(p.y + x.y);
        sum += w.z * FAST_TANH(p.z + x.z);
        sum += w.w * FAST_TANH(p.w + x.w);
    }

    // wave32 reduction
    #pragma unroll
    for (int off = 16; off > 0; off >>= 1)
        sum += __shfl_xor(sum, off, 32);

    if (lane == 0)
        energy[row] = (mask[row] == 0) ? -1000.0f : sum;
}

// ---------------------------------------------------------------------------
// Kernel 3: per-batch softmax over S=2048. One 256-thread block per b;
// 8 values/thread; wave shuffles + tiny LDS for cross-wave reduction.
// ---------------------------------------------------------------------------
__global__ __launch_bounds__(256)
void softmax_kernel(const float* __restrict__ energy,  // [B*S]
                    float* __restrict__ out)           // [B,1,S] flat
{
    __shared__ float red[8];
    const int b    = blockIdx.x;
    const int wave = threadIdx.x >> 5;
    const int lane = threadIdx.x & 31;
    const float* e = energy + (size_t)b * SQ;

    float v[8];
    float m = -INFINITY;
    #pragma unroll
    for (int i = 0; i < 8; ++i) {
        v[i] = e[threadIdx.x + i * 256];
        m = fmaxf(m, v[i]);
    }
    #pragma unroll
    for (int off = 16; off > 0; off >>= 1)
        m = fmaxf(m, __shfl_xor(m, off, 32));
    if (lane == 0) red[wave] = m;
    __syncthreads();
    m = red[0];
    #pragma unroll
    for (int i = 1; i < 8; ++i) m = fmaxf(m, red[i]);
    __syncthreads();

    float sum = 0.0f;
    #pragma unroll
    for (int i = 0; i < 8; ++i) {
        v[i] = __expf(v[i] - m);
        sum += v[i];
    }
    #pragma unroll
    for (int off = 16; off > 0; off >>= 1)
        sum += __shfl_xor(sum, off, 32);
    if (lane == 0) red[wave] = sum;
    __syncthreads();
    sum = 0.0f;
    #pragma unroll
    for (int i = 0; i < 8; ++i) sum += red[i];

    const float inv = 1.0f / sum;
    #pragma unroll
    for (int i = 0; i < 8; ++i)
        out[(size_t)b * SQ + threadIdx.x + i * 256] = v[i] * inv;
}

// ---------------------------------------------------------------------------
// Launch. Input order from setup_inputs():
//   0: query [1,B,Q] f32   1: projected_memory [B,S,Q] f32
//   2: mask [B,S] i32      3: Wq [Q,Q] f32   4: We [Q] f32
// Output: attn [B,1,S] f32. Workspace: pq (128KB) + energy (512KB).
// ---------------------------------------------------------------------------
extern "C" void kernel_launch(void* const* d_in, const int* in_sizes, int n_in,
                              void* d_out, int out_size, void* d_ws, size_t ws_size,
                              hipStream_t stream) {
    const float* query = (const float*)d_in[0];
    const float* pm    = (const float*)d_in[1];
    const int*   mask  = (const int*)  d_in[2];
    const float* Wq    = (const float*)d_in[3];
    const float* We    = (const float*)d_in[4];
    float*       out   = (float*)d_out;

    float* pq     = (float*)d_ws;                 // B*Q   = 32768 floats
    float* energy = pq + (size_t)BQ * QQ;         // B*S   = 131072 floats

    // 1) query projection via WMMA f32 16x16x4 (N-tiles x M-tiles)
    proj_query_wmma<<<dim3(QQ / 16, BQ / 16), 32, 0, stream>>>(query, Wq, pq);

    // 2) streaming tanh-dot + mask: 1 wave per (b,s) row, 8 rows per block
    energy_kernel<<<(BQ * SQ) / 8, 256, 0, stream>>>(pm, mask, pq, We, energy);

    // 3) per-batch softmax
    softmax_kernel<<<BQ, 256, 0, stream>>>(energy, out);
}